// TopKGate_57466662420617
// MI455X (gfx1250) — compile-verified
//
#include <hip/hip_runtime.h>
#include <math.h>

typedef __attribute__((ext_vector_type(2))) float v2f;
typedef __attribute__((ext_vector_type(8))) float v8f;

static constexpr int NEXP = 16;   // experts (and per-thread entry count in scan)
static constexpr int KC   = 64;   // K-chunk per LDS stage
static constexpr int LDSS = 68;   // 64 + 4 pad: conflict-free column reads, 16B-aligned rows

// ---------------------------------------------------------------------------
// Kernel 1: logits[N][16] = x[N][D] @ Wg[16][D]^T using V_WMMA_F32_16X16X4_F32
// grid: N/128 blocks x 256 threads (8 waves); each wave owns a 16-token tile.
// ---------------------------------------------------------------------------
__global__ __launch_bounds__(256)
void moe_gemm_wmma(const float* __restrict__ x, const float* __restrict__ Wg,
                   float* __restrict__ logits, int N, int D) {
    __shared__ float sA[8 * 16 * LDSS];   // per-wave 16x64 x-tiles
    __shared__ float sB[16 * LDSS];       // shared 16x64 Wg chunk

    const int tid     = threadIdx.x;
    const int wave    = tid >> 5;
    const int lane    = tid & 31;
    const int halfsel = lane >> 4;        // 0: lanes 0-15, 1: lanes 16-31
    const int l16     = lane & 15;
    const int tokenBase = (blockIdx.x * 8 + wave) * 16;

    float* myA = sA + wave * 16 * LDSS;

    v8f c = (v8f){0.f, 0.f, 0.f, 0.f, 0.f, 0.f, 0.f, 0.f};

    for (int k0 = 0; k0 < D; k0 += KC) {
        // Stage A: this wave's 16 tokens x 64 k values (256 float4 / 32 lanes).
        #pragma unroll
        for (int it = 0; it < 8; ++it) {
            int f   = it * 32 + lane;     // 0..255
            int row = f >> 4;             // token row 0..15
            int c4  = f & 15;             // float4 column
            const float4 v = *(const float4*)(x + (size_t)(tokenBase + row) * D + k0 + c4 * 4);
            *(float4*)(myA + row * LDSS + c4 * 4) = v;
        }
        // Stage B: Wg chunk 16 experts x 64 k (256 float4 / 256 threads).
        {
            int row = tid >> 4;
            int c4  = tid & 15;
            const float4 v = *(const float4*)(Wg + (size_t)row * D + k0 + c4 * 4);
            *(float4*)(sB + row * LDSS + c4 * 4) = v;
        }
        __syncthreads();

        // 16 WMMAs: A = 16x4 (M=l16; VGPR0 = K kk+2*halfsel, VGPR1 = +1)
        //           B = 4x16 (N=l16; same K striping) -> ds_load_b64 each.
        #pragma unroll
        for (int kk = 0; kk < KC; kk += 4) {
            const int koff = kk + halfsel * 2;
            const v2f a = *(const v2f*)(myA + l16 * LDSS + koff);
            const v2f b = *(const v2f*)(sB  + l16 * LDSS + koff);
            c = __builtin_amdgcn_wmma_f32_16x16x4_f32(false, a, false, b,
                                                      (short)0, c, false, false);
        }
        __syncthreads();
    }

    // C layout: VGPR v -> M = v + 8*halfsel, N = l16. Coalesced 64B stores.
    #pragma unroll
    for (int v = 0; v < 8; ++v) {
        int row = v + halfsel * 8;
        logits[(size_t)(tokenBase + row) * NEXP + l16] = c[v];
    }
}

// ---------------------------------------------------------------------------
// Kernel 2: softmax over 16 experts, top-2 (strict > = lowest-index ties),
// normalized weights, expert index stream, gate-sum accumulation for l_aux.
// ---------------------------------------------------------------------------
__global__ __launch_bounds__(256)
void moe_gate(const float* __restrict__ logits, int* __restrict__ idx_ws,
              float* __restrict__ weight_out, float* __restrict__ gate_sums, int N) {
    __shared__ float s_sum[NEXP];
    const int tid = threadIdx.x;
    const int t   = blockIdx.x * 256 + tid;
    if (tid < NEXP) s_sum[tid] = 0.f;
    __syncthreads();

    float g[NEXP];
    #pragma unroll
    for (int e = 0; e < NEXP; ++e) g[e] = logits[(size_t)t * NEXP + e];

    float m = g[0];
    #pragma unroll
    for (int e = 1; e < NEXP; ++e) m = fmaxf(m, g[e]);
    float s = 0.f;
    #pragma unroll
    for (int e = 0; e < NEXP; ++e) { g[e] = expf(g[e] - m); s += g[e]; }
    float inv = 1.f / s;
    #pragma unroll
    for (int e = 0; e < NEXP; ++e) g[e] *= inv;

    float v1 = -1.f; int i1 = 0; float v2 = -1.f; int i2 = 0;
    #pragma unroll
    for (int e = 0; e < NEXP; ++e) {
        float ge = g[e];
        if (ge > v1)      { v2 = v1; i2 = i1; v1 = ge; i1 = e; }
        else if (ge > v2) { v2 = ge; i2 = e; }
    }

    idx_ws[t]     = i1;
    idx_ws[N + t] = i2;
    float wn = 1.f / (v1 + v2);
    weight_out[t]     = v1 * wn;
    weight_out[N + t] = v2 * wn;

    #pragma unroll
    for (int e = 0; e < NEXP; ++e) atomicAdd(&s_sum[e], g[e]);
    __syncthreads();
    if (tid < NEXP) atomicAdd(&gate_sums[tid], s_sum[tid]);
}

// ---------------------------------------------------------------------------
// Kernel 3: exact ordered scan (cumsum-over-mask) for positions, counts,
// offsets, scatter, and l_aux. One block, 1024 threads, 16 entries/thread.
// ---------------------------------------------------------------------------
__global__ __launch_bounds__(1024)
void moe_scan(const int* __restrict__ idx_ws, const float* __restrict__ gate_sums,
              float* __restrict__ l_aux_out, float* __restrict__ pos_after_out,
              float* __restrict__ pos_before_out, float* __restrict__ count_out, int N) {
    __shared__ int hist[1024][NEXP];   // 64 KB
    __shared__ int expOff[NEXP];
    __shared__ int top1cnt[NEXP];
    __shared__ int totals[NEXP];

    const int t    = threadIdx.x;
    const int base = t * NEXP;                 // 16 entries per thread
    const int halfThread = (N / NEXP) - 1;     // last thread covering top-1 half

    int myidx[NEXP];
    #pragma unroll
    for (int e = 0; e < NEXP; ++e) hist[t][e] = 0;
    #pragma unroll
    for (int j = 0; j < NEXP; ++j) {
        int e = idx_ws[base + j];
        myidx[j] = e;
        hist[t][e]++;
    }
    __syncthreads();

    int h[NEXP], r[NEXP];
    #pragma unroll
    for (int e = 0; e < NEXP; ++e) { h[e] = hist[t][e]; r[e] = h[e]; }

    // Hillis-Steele inclusive scan of 16-wide vectors across 1024 threads.
    for (int d = 1; d < 1024; d <<= 1) {
        int tmp[NEXP];
        #pragma unroll
        for (int e = 0; e < NEXP; ++e) tmp[e] = (t >= d) ? hist[t - d][e] : 0;
        __syncthreads();
        #pragma unroll
        for (int e = 0; e < NEXP; ++e) { r[e] += tmp[e]; hist[t][e] = r[e]; }
        __syncthreads();
    }

    if (t == halfThread) {      // inclusive over entries [0, N) = top-1 counts
        #pragma unroll
        for (int e = 0; e < NEXP; ++e) top1cnt[e] = r[e];
    }
    if (t == 1023) {
        #pragma unroll
        for (int e = 0; e < NEXP; ++e) totals[e] = r[e];
    }
    __syncthreads();

    if (t == 0) {
        int off = 0;
        for (int e = 0; e < NEXP; ++e) { expOff[e] = off; off += totals[e]; }
    }
    // Repurpose own hist row as running (exclusive) counters.
    #pragma unroll
    for (int e = 0; e < NEXP; ++e) hist[t][e] = r[e] - h[e];
    __syncthreads();

    #pragma unroll
    for (int j = 0; j < NEXP; ++j) {
        int e   = myidx[j];
        int pos = hist[t][e]++;
        int pa  = pos + expOff[e];
        pos_after_out[base + j] = (float)pa;
        pos_before_out[pa]      = (float)(base + j);   // permutation: full coverage
    }

    if (t < NEXP) count_out[t] = (float)totals[t];
    if (t == 0) {
        float invN = 1.f / (float)N;
        float laux = 0.f;
        for (int e = 0; e < NEXP; ++e)
            laux += (gate_sums[e] * invN) * ((float)top1cnt[e] * invN);
        l_aux_out[0] = laux * (float)NEXP;
    }
}

// ---------------------------------------------------------------------------
extern "C" void kernel_launch(void* const* d_in, const int* in_sizes, int n_in,
                              void* d_out, int out_size, void* d_ws, size_t ws_size,
                              hipStream_t stream) {
    const float* x  = (const float*)d_in[0];
    const float* Wg = (const float*)d_in[1];
    const int D = in_sizes[1] / NEXP;   // 4096
    const int N = in_sizes[0] / D;      // 8192

    char*  ws        = (char*)d_ws;
    float* logits    = (float*)ws;                                         // N*16 floats
    int*   idx_ws    = (int*)(ws + (size_t)N * NEXP * sizeof(float));      // 2N ints
    float* gate_sums = (float*)(ws + (size_t)N * NEXP * sizeof(float)
                                   + (size_t)2 * N * sizeof(int));         // 16 floats

    // Output tuple, flattened: l_aux | pos_after[2N] | pos_before[2N] | counts[16] | weight[2N]
    float* out            = (float*)d_out;
    float* l_aux_out      = out;
    float* pos_after_out  = out + 1;
    float* pos_before_out = out + 1 + 2 * N;
    float* count_out      = out + 1 + 4 * N;
    float* weight_out     = out + 1 + 4 * N + NEXP;

    hipMemsetAsync(gate_sums, 0, NEXP * sizeof(float), stream);

    moe_gemm_wmma<<<dim3(N / 128), dim3(256), 0, stream>>>(x, Wg, logits, N, D);
    moe_gate<<<dim3(N / 256), dim3(256), 0, stream>>>(logits, idx_ws, weight_out,
                                                      gate_sums, N);
    moe_scan<<<dim3(1), dim3(1024), 0, stream>>>(idx_ws, gate_sums, l_aux_out,
                                                 pos_after_out, pos_before_out,
                                                 count_out, N);
}